// ImplicitValueAttention_63445256896984
// MI455X (gfx1250) — compile-verified
//
#include <hip/hip_runtime.h>
#include <hip/hip_bf16.h>

// Problem constants (fixed by the reference).
#define BATCH 2
#define SEQ   2048
#define DMODEL 1024
#define NHEAD 16
#define HD    64
#define BN    (BATCH * SEQ)   // 4096 rows

typedef __attribute__((ext_vector_type(2))) float v2f;
typedef __attribute__((ext_vector_type(8))) float v8f;

#define ACT_NONE 0
#define ACT_SILU 1

// CDNA5 async global->LDS path (guarded: falls back to reg-prefetch pipeline).
#if defined(__gfx1250__) && __has_builtin(__builtin_amdgcn_global_load_async_to_lds_b32) && \
    __has_builtin(__builtin_amdgcn_s_wait_asynccnt)
#define USE_ASYNC_LDS 1
#else
#define USE_ASYNC_LDS 0
#endif

#if USE_ASYNC_LDS
typedef __attribute__((address_space(1))) int* gptr_i32;
typedef __attribute__((address_space(3))) int* lptr_i32;
__device__ __forceinline__ void async_load_f32_to_lds(const float* g, float* l) {
  // Emits GLOBAL_LOAD_ASYNC_TO_LDS_B32 (tracked by ASYNCcnt).
  __builtin_amdgcn_global_load_async_to_lds_b32((gptr_i32)g, (lptr_i32)l, 0, 0);
}
#endif

// ---------------------------------------------------------------------------
// FP32 WMMA GEMM: C[M,N] = act(A[M,K] @ W[K,N]),  N/K compile-time constants.
// Register-blocked: each wave computes a 64x32 macro-tile = 4x2 WMMA tiles.
// Inner K-step (K+=4): 4 x b64 A-loads + 4 x b32 B-loads feed 8 WMMAs
// (1 VMEM op per v_wmma_f32_16x16x4_f32).
//
// Fragment layouts per CDNA5 ISA 7.12.2 (wave32):
//   A 16x4:  lanes 0-15 -> (M=lane, K=0..1); lanes 16-31 -> (M=lane-16, K=2..3)
//   B 4x16:  lanes 0-15 -> (K=0..1, N=lane); lanes 16-31 -> (K=2..3, N=lane-16)
//   C 16x16: v[j]: lanes 0-15 -> (M=j, N=lane); lanes 16-31 -> (M=j+8, N=lane-16)
// ---------------------------------------------------------------------------
template <int ACT, int N, int K>
__global__ __launch_bounds__(256) void gemm_f32_wmma(
    const float* __restrict__ A, const float* __restrict__ W,
    float* __restrict__ C) {
  const int lane = threadIdx.x & 31;
  const int wave = threadIdx.x >> 5;
  const int m0 = blockIdx.y * 64;                // wave M-tile
  const int n0 = blockIdx.x * 256 + wave * 32;   // wave N-tile
  const int half = lane >> 4;                    // 0: lanes 0-15, 1: lanes 16-31
  const int l16 = lane & 15;

  v8f acc[4][2] = {};

  const float* aptr[4];
#pragma unroll
  for (int mi = 0; mi < 4; ++mi)
    aptr[mi] = A + (size_t)(m0 + mi * 16 + l16) * K + half * 2;
  // B base: row (half*2), column n0+l16; advances by k*N each step.
  const float* bbase = W + (size_t)(half * 2) * N + n0 + l16;

#pragma unroll 2
  for (int k = 0; k < K; k += 4) {
    v2f af[4], bf[2];
#pragma unroll
    for (int mi = 0; mi < 4; ++mi)
      af[mi] = *(const v2f*)(aptr[mi] + k);       // 8B aligned (k even)
#pragma unroll
    for (int ni = 0; ni < 2; ++ni) {
      const float* bp = bbase + (size_t)k * N + ni * 16;
      bf[ni].x = bp[0];       // K = k + half*2
      bf[ni].y = bp[N];       // K = k + half*2 + 1
    }
#pragma unroll
    for (int mi = 0; mi < 4; ++mi)
#pragma unroll
      for (int ni = 0; ni < 2; ++ni)
        acc[mi][ni] = __builtin_amdgcn_wmma_f32_16x16x4_f32(
            false, af[mi], false, bf[ni], (short)0, acc[mi][ni], false, false);
  }

#pragma unroll
  for (int mi = 0; mi < 4; ++mi)
#pragma unroll
    for (int ni = 0; ni < 2; ++ni)
#pragma unroll
      for (int j = 0; j < 8; ++j) {
        float val = acc[mi][ni][j];
        if (ACT == ACT_SILU) val = val / (1.0f + __expf(-val));
        C[(size_t)(m0 + mi * 16 + half * 8 + j) * N + (n0 + ni * 16 + l16)] = val;
      }
}

// ---------------------------------------------------------------------------
// Gates: beta = 2*sigmoid(x @ Wbet), a = sigmoid(x @ Wf).  Output [BN, H] each.
// ---------------------------------------------------------------------------
__global__ __launch_bounds__(256) void gates_kernel(
    const float* __restrict__ x, const float* __restrict__ Wbet,
    const float* __restrict__ Wf, float* __restrict__ beta_o,
    float* __restrict__ a_o) {
  const int id = blockIdx.x * 256 + threadIdx.x;
  if (id >= BN * NHEAD * 2) return;
  const int m = id >> 5;        // row (all 32 lanes of a wave share it)
  const int rem = id & 31;
  const int hh = rem & 15;
  const int which = rem >> 4;   // 0 -> beta, 1 -> a
  const float* __restrict__ Wm = which ? Wf : Wbet;
  const float* __restrict__ xr = x + (size_t)m * DMODEL;
  float acc = 0.f;
  for (int kk = 0; kk < DMODEL; ++kk)
    acc = fmaf(xr[kk], Wm[kk * NHEAD + hh], acc);
  const float s = 1.0f / (1.0f + __expf(-acc));
  if (which) a_o[m * NHEAD + hh] = s;
  else       beta_o[m * NHEAD + hh] = 2.0f * s;
}

// ---------------------------------------------------------------------------
// Gated delta-rule scan, one workgroup per (batch, head), 64 threads.
// Thread t owns column t of state S[64][64] in VGPRs.
// Software-pipelined: step t+1's k/q rows stream into the other half of a
// double-buffered LDS stage while step t computes (async global->LDS on
// CDNA5; register prefetch + ds_store fallback otherwise).
// Fuses: k l2-norm, grouped RMSNorm over hd, norm_w scaling.
// ---------------------------------------------------------------------------
__global__ __launch_bounds__(64) void scan_kernel(
    const float* __restrict__ q, const float* __restrict__ k,
    const float* __restrict__ v, const float* __restrict__ beta,
    const float* __restrict__ a, const float* __restrict__ norm_w,
    float* __restrict__ o) {
  const int bh = blockIdx.x;    // 0..31
  const int bb = bh >> 4;
  const int hh = bh & 15;
  const int t = threadIdx.x;    // owned value-dim column, 0..63

  __shared__ float sk[2][HD];
  __shared__ float sq[2][HD];
  __shared__ float so[HD];

  float S[HD];
#pragma unroll
  for (int i = 0; i < HD; ++i) S[i] = 0.f;
  const float nw = norm_w[hh * HD + t];

  const size_t row0 = (size_t)bb * SEQ;
  const size_t head_off = (size_t)hh * HD + t;

  // Prime buffer 0 with step 0 synchronously.
  sk[0][t] = k[row0 * DMODEL + head_off];
  sq[0][t] = q[row0 * DMODEL + head_off];

  int buf = 0;
  for (int step = 0; step < SEQ; ++step) {
    const size_t row = row0 + step;
    const size_t base = row * DMODEL + head_off;
    const int nxt = buf ^ 1;

    // ---- issue next step's k/q prefetch into the other buffer ----
#if USE_ASYNC_LDS
    if (step + 1 < SEQ) {
      async_load_f32_to_lds(k + base + DMODEL, &sk[nxt][t]);
      async_load_f32_to_lds(q + base + DMODEL, &sq[nxt][t]);
    }
#else
    float knext = 0.f, qnext = 0.f;
    if (step + 1 < SEQ) {
      knext = k[base + DMODEL];
      qnext = q[base + DMODEL];
    }
#endif

    const float vv = v[base];
    const float at = a[row * NHEAD + hh];
    const float bt = beta[row * NHEAD + hh];

#if USE_ASYNC_LDS
    // Oldest 2 async loads (previous iteration's prefetch) must have landed;
    // the 2 just issued stay in flight. Async loads complete in order.
    __builtin_amdgcn_s_wait_asynccnt(2);
#endif
    __syncthreads();  // barrier1: current buffer fully visible

    const float* __restrict__ ck = sk[buf];
    const float* __restrict__ cq = sq[buf];

    // l2 normalization factor of k.
    float ss = 1e-12f;
#pragma unroll
    for (int i = 0; i < HD; ++i) ss = fmaf(ck[i], ck[i], ss);
    const float kinv = rsqrtf(ss);

    // u[t] = v[t] - a * (kn . S_col)
    float dot = 0.f;
#pragma unroll
    for (int i = 0; i < HD; ++i) dot = fmaf(ck[i], S[i], dot);
    const float u = vv - at * dot * kinv;

    // State update + output projection in one pass.
    const float bu = bt * u * kinv;
    float oacc = 0.f;
#pragma unroll
    for (int i = 0; i < HD; ++i) {
      S[i] = fmaf(at, S[i], bu * ck[i]);
      oacc = fmaf(cq[i], S[i], oacc);
    }

#if !USE_ASYNC_LDS
    // Safe: after barrier1(step), all step-1 readers of buf^1 are done.
    if (step + 1 < SEQ) {
      sk[nxt][t] = knext;
      sq[nxt][t] = qnext;
    }
#endif
    so[t] = oacc;
    __syncthreads();  // barrier2

    // Grouped RMSNorm over the head dim + norm_w.
    float ms = 0.f;
#pragma unroll
    for (int i = 0; i < HD; ++i) ms = fmaf(so[i], so[i], ms);
    const float scale = rsqrtf(ms * (1.0f / HD) + 1e-6f);
    o[base] = oacc * scale * nw;

    buf = nxt;
  }
}

// ---------------------------------------------------------------------------
// Launcher.  Inputs: x, Wq, Wk, Wv, Wo, Wbet, Wf, norm_w (all fp32).
// ---------------------------------------------------------------------------
extern "C" void kernel_launch(void* const* d_in, const int* in_sizes, int n_in,
                              void* d_out, int out_size, void* d_ws,
                              size_t ws_size, hipStream_t stream) {
  const float* x      = (const float*)d_in[0];
  const float* Wq     = (const float*)d_in[1];
  const float* Wk     = (const float*)d_in[2];
  const float* Wv     = (const float*)d_in[3];
  const float* Wo     = (const float*)d_in[4];
  const float* Wbet   = (const float*)d_in[5];
  const float* Wf     = (const float*)d_in[6];
  const float* norm_w = (const float*)d_in[7];

  float* ws = (float*)d_ws;
  const size_t big = (size_t)BN * DMODEL;   // 4M floats
  float* qb = ws;               // silu(x@Wq)
  float* kb = qb + big;         // silu(x@Wk)
  float* vb = kb + big;         // x@Wv
  float* ob = vb + big;         // scan output (already rms-normed * norm_w)
  float* betab = ob + big;      // [BN,H]
  float* ab = betab + (size_t)BN * NHEAD;

  dim3 gemm_grid(DMODEL / 256, BN / 64);    // (4, 64) blocks, 8 waves each

  gemm_f32_wmma<ACT_SILU, DMODEL, DMODEL><<<gemm_grid, 256, 0, stream>>>(x, Wq, qb);
  gemm_f32_wmma<ACT_SILU, DMODEL, DMODEL><<<gemm_grid, 256, 0, stream>>>(x, Wk, kb);
  gemm_f32_wmma<ACT_NONE, DMODEL, DMODEL><<<gemm_grid, 256, 0, stream>>>(x, Wv, vb);

  gates_kernel<<<(BN * NHEAD * 2 + 255) / 256, 256, 0, stream>>>(x, Wbet, Wf, betab, ab);

  scan_kernel<<<BATCH * NHEAD, 64, 0, stream>>>(qb, kb, vb, betab, ab, norm_w, ob);

  gemm_f32_wmma<ACT_NONE, DMODEL, DMODEL><<<gemm_grid, 256, 0, stream>>>(ob, Wo, (float*)d_out);
}